// CSNNet_88880053223709
// MI455X (gfx1250) — compile-verified
//
#include <hip/hip_runtime.h>

// ---------------------------------------------------------------------------
// Spiking CNN forward, MI455X (gfx1250, wave32).
//
// mem1 (32 x 524288 f32) recurrence is elementwise-independent; keep all 50
// steps of state in VGPRs. Per step, spk1 @ fc_w.T [524288x2] is reduced with
// V_WMMA_F32_16X16X4_F32 (M=16 batch rows, K=4 js, N cols 0..1 = fc_w rows,
// rest zero). Conv+pool fused into the prologue. Global traffic: x (4MiB,
// L2-resident), fc_w (4MiB, once), per-block partials (~13MB).
//
// v2: reset[t] == spk[t-1] (both are (mem1 > THR) of the same value), so the
// previous spike fragment is kept in registers instead of recomputing the
// reset compare+select (-2 VALU/element/step). Two WMMA accumulators break
// the serial D->C chain for better WMMA/VALU co-execution.
// ---------------------------------------------------------------------------

typedef float v2f __attribute__((ext_vector_type(2)));
typedef float v8f __attribute__((ext_vector_type(8)));

#define STEPS   50
#define BATCH   32
#define INPUT   32768
#define POOLED  16384
#define J_TOTAL (32 * 16384)      // 524288 flattened (channel, pooled) features
#define BETA    0.95f
#define THR     1.0f

#define JPW          64                       // j's owned per wave
#define CHUNKS       (JPW / 4)                // 16 WMMA K-chunks per step
#define THREADS_MAIN 512
#define WAVES_MAIN   (THREADS_MAIN / 32)      // 16 waves / block
#define TOTAL_WAVES  ((J_TOTAL / JPW) * 2)    // 8192 chunks x 2 M-tiles = 16384
#define NBLOCKS_MAIN (TOTAL_WAVES / WAVES_MAIN) // 1024

// ---------------------------------------------------------------------------
// Main kernel: fused conv+pool prologue, 50-step leaky recurrence in
// registers, WMMA f32 16x16x4 reduction, deterministic LDS block reduction.
// partial layout: [STEPS][NBLOCKS_MAIN][64]  (64 = batch*2 outputs)
// ---------------------------------------------------------------------------
__global__ __launch_bounds__(THREADS_MAIN)
void snn_main(const float* __restrict__ x,      // [32][1][32768]
              const float* __restrict__ convw,  // [32][1][3]
              const float* __restrict__ convb,  // [32]
              const float* __restrict__ fcw,    // [2][524288]
              float* __restrict__ partial)
{
    __shared__ float lwv[WAVES_MAIN][64];

    const int tid   = threadIdx.x;
    const int lane  = tid & 31;
    const int wv    = tid >> 5;
    const int gw    = blockIdx.x * WAVES_MAIN + wv;
    const int mtile = gw & 1;          // batch tile: rows 0..15 or 16..31
    const int chunk = gw >> 1;         // which 64-j chunk
    const int j0    = chunk * JPW;

    // WMMA lane roles (ISA 16x16x4 f32 layouts):
    //   A: lane l, vgpr v -> (M = l&15, K = 2*(l>>4)+v)
    //   B: lane l, vgpr v -> (K = 2*(l>>4)+v, N = l&15)
    //   D: lane l, vgpr d -> (M = d + 8*(l>>4), N = l&15)
    const int mrow = lane & 15;
    const int kh   = lane >> 4;
    const int n    = lane & 15;        // B/D column this lane carries
    const int b    = mtile * 16 + mrow;

    // j = c*POOLED + p ; a 64-aligned chunk never crosses a channel.
    const int c  = j0 >> 14;
    const int p0 = j0 & (POOLED - 1);

    const float w0 = convw[c * 3 + 0];
    const float w1 = convw[c * 3 + 1];
    const float w2 = convw[c * 3 + 2];
    const float bias = convb[c];
    const float* __restrict__ xb = x + (size_t)b * INPUT;

    // ---- fused conv1d(pad=1) + maxpool(k=2): loop-invariant drive current --
    v2f cur[CHUNKS];
#pragma unroll
    for (int kk = 0; kk < CHUNKS; ++kk) {
#pragma unroll
        for (int v = 0; v < 2; ++v) {
            const int p  = p0 + 4 * kk + 2 * kh + v;
            const int q0 = 2 * p;
            const int q1 = 2 * p + 1;
            const float xm1 = (q0 == 0)         ? 0.0f : xb[q0 - 1];
            const float x0  = xb[q0];
            const float x1  = xb[q1];
            const float x2  = (q1 == INPUT - 1) ? 0.0f : xb[q1 + 1];
            const float c0 = bias + w0 * xm1 + w1 * x0 + w2 * x1;
            const float c1 = bias + w0 * x0  + w1 * x1 + w2 * x2;
            cur[kk][v] = fmaxf(c0, c1);
        }
    }

    // ---- loop-invariant B fragments: fc_w rows in N columns 0..1 ----------
    v2f Bv[CHUNKS];
#pragma unroll
    for (int kk = 0; kk < CHUNKS; ++kk) {
#pragma unroll
        for (int v = 0; v < 2; ++v) {
            const int k = 2 * kh + v;
            const int j = j0 + 4 * kk + k;
            Bv[kk][v] = (n < 2) ? fcw[(size_t)n * J_TOTAL + j] : 0.0f;
        }
    }

    // ---- membrane state + previous-spike (== reset indicator) in VGPRs ----
    v2f mem[CHUNKS];
    v2f spk[CHUNKS];
#pragma unroll
    for (int kk = 0; kk < CHUNKS; ++kk) {
        mem[kk][0] = 0.0f; mem[kk][1] = 0.0f;
        spk[kk][0] = 0.0f; spk[kk][1] = 0.0f;
    }

    for (int t = 0; t < STEPS; ++t) {
        v8f d0 = {};   // two accumulators: halve the WMMA D->C serial chain
        v8f d1 = {};
#pragma unroll
        for (int kk = 0; kk < CHUNKS; ++kk) {
            v2f a;
#pragma unroll
            for (int v = 0; v < 2; ++v) {
                // reset*THR == previous spike (THR==1): subtract-reset leaky
                const float m = fmaf(BETA, mem[kk][v], cur[kk][v]) - spk[kk][v];
                mem[kk][v] = m;
                const float s = (m > THR) ? 1.0f : 0.0f;
                spk[kk][v] = s;
                a[v] = s;
            }
            if (kk & 1)
                d1 = __builtin_amdgcn_wmma_f32_16x16x4_f32(
                    false, a, false, Bv[kk], (short)0, d1, false, false);
            else
                d0 = __builtin_amdgcn_wmma_f32_16x16x4_f32(
                    false, a, false, Bv[kk], (short)0, d0, false, false);
        }

        // ---- deterministic block reduction (no atomics) -------------------
        __syncthreads();                       // protect previous iter's reads
        {   // zero all 16*64 slots (512 threads -> 2 each)
            ((float*)lwv)[tid]       = 0.0f;
            ((float*)lwv)[tid + 512] = 0.0f;
        }
        __syncthreads();
        if (n < 2) {
            // each (wave, slot) written by exactly one lane: deterministic
#pragma unroll
            for (int d = 0; d < 8; ++d) {
                const int brow = mtile * 16 + d + 8 * kh;       // batch row
                lwv[wv][brow * 2 + n] = d0[d] + d1[d];
            }
        }
        __syncthreads();
        if (tid < 64) {
            float s = 0.0f;
#pragma unroll
            for (int w = 0; w < WAVES_MAIN; ++w) s += lwv[w][tid];
            partial[((size_t)t * NBLOCKS_MAIN + blockIdx.x) * 64 + tid] = s;
        }
    }
}

// ---------------------------------------------------------------------------
// Reduce block partials: one block per timestep; coalesced, deterministic.
// red[t*64 + b*2 + o] = sum_blocks partial + fc_b[o]
// ---------------------------------------------------------------------------
__global__ __launch_bounds__(256)
void snn_reduce(const float* __restrict__ partial,
                const float* __restrict__ fcb,
                float* __restrict__ red)
{
    __shared__ float ls[4][64];
    const int t  = blockIdx.x;
    const int i  = threadIdx.x & 63;
    const int sl = threadIdx.x >> 6;
    float s = 0.0f;
    for (int blk = sl; blk < NBLOCKS_MAIN; blk += 4)
        s += partial[((size_t)t * NBLOCKS_MAIN + blk) * 64 + i];
    ls[sl][i] = s;
    __syncthreads();
    if (threadIdx.x < 64)
        red[t * 64 + i] = ls[0][i] + ls[1][i] + ls[2][i] + ls[3][i] + fcb[i & 1];
}

// ---------------------------------------------------------------------------
// Output-layer leaky neuron (64 states, sequential over 50 steps) + emit
// spk_rec [50][32][2] then mem_rec [50][32][2].
// ---------------------------------------------------------------------------
__global__ __launch_bounds__(64)
void snn_finalize(const float* __restrict__ red, float* __restrict__ out)
{
    const int i = threadIdx.x;     // i = b*2 + o
    float memo = 0.0f;
    for (int t = 0; t < STEPS; ++t) {
        const float curo = red[t * 64 + i];
        const float r = (memo > THR) ? THR : 0.0f;
        memo = BETA * memo + curo - r;
        const float spk = (memo > THR) ? 1.0f : 0.0f;
        out[t * 64 + i] = spk;                       // spk_rec
        out[STEPS * 64 + t * 64 + i] = memo;         // mem_rec
    }
}

// ---------------------------------------------------------------------------
extern "C" void kernel_launch(void* const* d_in, const int* in_sizes, int n_in,
                              void* d_out, int out_size, void* d_ws, size_t ws_size,
                              hipStream_t stream)
{
    const float* x     = (const float*)d_in[0];
    const float* convw = (const float*)d_in[1];
    const float* convb = (const float*)d_in[2];
    const float* fcw   = (const float*)d_in[3];
    const float* fcb   = (const float*)d_in[4];
    float* out = (float*)d_out;

    float* partial = (float*)d_ws;                                   // 13.1 MB
    float* red = (float*)((char*)d_ws +
                 (size_t)STEPS * NBLOCKS_MAIN * 64 * sizeof(float)); // 12.8 KB

    snn_main<<<NBLOCKS_MAIN, THREADS_MAIN, 0, stream>>>(x, convw, convb, fcw, partial);
    snn_reduce<<<STEPS, 256, 0, stream>>>(partial, fcb, red);
    snn_finalize<<<1, 64, 0, stream>>>(red, out);
}